// GNNOnly_71683004170810
// MI455X (gfx1250) — compile-verified
//
#include <hip/hip_runtime.h>
#include <math.h>

#define Nn 50000
#define Ee 800000
#define Hd 128
#define Ld 3
#define Gd 512

typedef __attribute__((ext_vector_type(2))) float v2f;
typedef __attribute__((ext_vector_type(8))) float v8f;

// ---------------- utility ----------------
__global__ __launch_bounds__(256) void k_zero(float* __restrict__ p, int n) {
  int i = blockIdx.x * 256 + threadIdx.x;
  if (i < n) p[i] = 0.0f;
}

// out[n,128] = x[n,indim] @ w[indim,128] + b
__global__ __launch_bounds__(256) void k_proj(const float* __restrict__ x,
                                              const float* __restrict__ w,
                                              const float* __restrict__ b,
                                              float* __restrict__ out,
                                              int indim) {
  int idx = blockIdx.x * 256 + threadIdx.x;
  if (idx >= Nn * Hd) return;
  int n = idx >> 7, j = idx & (Hd - 1);
  float s = b[j];
  for (int k = 0; k < indim; ++k) s = fmaf(x[n * indim + k], w[k * Hd + j], s);
  out[idx] = s;
}

// ---------------- WMMA f32 GEMM: OUT[N,128] = A[N,128] @ W[128,128] + bias ----------------
// block = 256 threads = 8 waves; block covers 64 rows x 128 cols.
// wave w: rowTile = w>>1 (16 rows), colBase = (w&1)*64 (4 col tiles of 16).
__global__ __launch_bounds__(256) void k_gemm(const float* __restrict__ A,
                                              const float* __restrict__ W,
                                              const float* __restrict__ bias,
                                              float* __restrict__ OUT) {
  __shared__ float ldsA[64 * Hd];
  const int tid  = threadIdx.x;
  const int lane = tid & 31;
  const int wv   = tid >> 5;
  const int wgRow = blockIdx.x * 64;

  // stage A block [64 x 128] into LDS (clamp tail rows; they are computed but never stored)
  for (int i = tid; i < 64 * (Hd / 4); i += 256) {
    int r  = i >> 5;
    int c4 = (i & 31) << 2;
    int gr = wgRow + r;
    if (gr >= Nn) gr = Nn - 1;
    *(float4*)(&ldsA[r * Hd + c4]) = *(const float4*)(A + (size_t)gr * Hd + c4);
  }
  __syncthreads();

  const int rowTile = wv >> 1;
  const int colBase = (wv & 1) * 64;
  const int m  = lane & 15;   // row index within A tile / col index within B,C,D tiles
  const int kh = lane >> 4;   // K-half select per ISA v2f layout

  v8f acc[4] = {};
  const float* arow = &ldsA[(rowTile * 16 + m) * Hd];

  for (int kb = 0; kb < Hd; kb += 4) {
    const int k0 = kb + 2 * kh;
    // A 16x4 f32 layout: lane m, vgpr0 = A[m][k0], vgpr1 = A[m][k0+1]
    v2f a;
    a.x = arow[k0];
    a.y = arow[k0 + 1];
    const float* w0 = W + (size_t)k0 * Hd + colBase + m;  // row k0 of W
#pragma unroll
    for (int t = 0; t < 4; ++t) {
      // B 4x16 f32 layout: lane n=m, vgpr0 = B[k0][n], vgpr1 = B[k0+1][n]
      v2f b;
      b.x = w0[t * 16];
      b.y = w0[Hd + t * 16];
      acc[t] = __builtin_amdgcn_wmma_f32_16x16x4_f32(
          false, a, false, b, (short)0, acc[t], false, false);
    }
  }

  // bias + store. D layout: lane L, vgpr v -> row (v + 8*(L>=16)), col L%16
#pragma unroll
  for (int t = 0; t < 4; ++t) {
    const int cc = colBase + t * 16 + m;
    const float bv = bias[cc];
#pragma unroll
    for (int v = 0; v < 8; ++v) {
      int rr = wgRow + rowTile * 16 + v + 8 * kh;
      if (rr < Nn) OUT[(size_t)rr * Hd + cc] = acc[t][v] + bv;
    }
  }
}

// ---------------- edge degree counts ----------------
__global__ __launch_bounds__(256) void k_count(const int* __restrict__ row,
                                               float* __restrict__ cnt) {
  int e = blockIdx.x * 256 + threadIdx.x;
  if (e < Ee) atomicAdd(&cnt[row[e]], 1.0f);
}

// ---------------- gather y[col] and scatter-add into agg[row] ----------------
// 32 threads per edge, float4 per thread (128 floats/edge)
__global__ __launch_bounds__(256) void k_scatter(const float* __restrict__ y,
                                                 const int* __restrict__ row,
                                                 const int* __restrict__ col,
                                                 float* __restrict__ agg) {
  int gid = blockIdx.x * 256 + threadIdx.x;
  if (gid >= Ee * 32) return;
  int e = gid >> 5;
  int q = (gid & 31) << 2;
  int c = col[e];
  int r = row[e];
  const float4 v = *(const float4*)(y + (size_t)c * Hd + q);
  float* dst = agg + (size_t)r * Hd + q;
  atomicAdd(dst + 0, v.x);
  atomicAdd(dst + 1, v.y);
  atomicAdd(dst + 2, v.z);
  atomicAdd(dst + 3, v.w);
}

// ---------------- fused: s = self + agg/max(cnt,1); h += relu(LN(s)*g+b) ----------------
// one wave32 per node; 4 H-elements per lane
__global__ __launch_bounds__(256) void k_update(float* __restrict__ h,
                                                const float* __restrict__ selfo,
                                                const float* __restrict__ agg,
                                                const float* __restrict__ cnt,
                                                const float* __restrict__ g,
                                                const float* __restrict__ b) {
  const int node = blockIdx.x * 8 + (threadIdx.x >> 5);
  const int lane = threadIdx.x & 31;
  if (node >= Nn) return;
  const size_t base = (size_t)node * Hd + lane * 4;
  const float4 s4 = *(const float4*)(selfo + base);
  const float4 a4 = *(const float4*)(agg + base);
  const float inv = 1.0f / fmaxf(cnt[node], 1.0f);
  float x0 = s4.x + a4.x * inv;
  float x1 = s4.y + a4.y * inv;
  float x2 = s4.z + a4.z * inv;
  float x3 = s4.w + a4.w * inv;
  float sum = x0 + x1 + x2 + x3;
  for (int off = 16; off; off >>= 1) sum += __shfl_xor(sum, off, 32);
  const float mean = sum * (1.0f / Hd);
  const float d0 = x0 - mean, d1 = x1 - mean, d2 = x2 - mean, d3 = x3 - mean;
  float vs = d0 * d0 + d1 * d1 + d2 * d2 + d3 * d3;
  for (int off = 16; off; off >>= 1) vs += __shfl_xor(vs, off, 32);
  const float r = rsqrtf(vs * (1.0f / Hd) + 1e-5f);
  const float4 g4 = *(const float4*)(g + lane * 4);
  const float4 b4 = *(const float4*)(b + lane * 4);
  float4 h4 = *(const float4*)(h + base);
  h4.x += fmaxf(fmaf(d0 * r, g4.x, b4.x), 0.0f);
  h4.y += fmaxf(fmaf(d1 * r, g4.y, b4.y), 0.0f);
  h4.z += fmaxf(fmaf(d2 * r, g4.z, b4.z), 0.0f);
  h4.w += fmaxf(fmaf(d3 * r, g4.w, b4.w), 0.0f);
  *(float4*)(h + base) = h4;
}

// ---------------- mean pool (partial sums + few atomics; divide later) ----------------
__global__ __launch_bounds__(128) void k_pool(const float* __restrict__ h,
                                              float* __restrict__ out) {
  const int col = threadIdx.x;
  int r0 = blockIdx.x * 512;
  int r1 = r0 + 512;
  if (r1 > Nn) r1 = Nn;
  float s = 0.0f;
  for (int r = r0; r < r1; ++r) s += h[(size_t)r * Hd + col];
  atomicAdd(&out[col], s);
}

// ---------------- fusion + head + K/delay/rho; writes all 1793 outputs ----------------
// out layout: u[1024] | rho[1] | K[512] | hE[128] | hI[128]
__global__ __launch_bounds__(256) void k_final(const float* __restrict__ pooled,
                                               const float* __restrict__ fw,
                                               const float* __restrict__ fb,
                                               const float* __restrict__ hw,
                                               const float* __restrict__ hb,
                                               const float* __restrict__ logK,
                                               const float* __restrict__ tau,
                                               const float* __restrict__ tau_max,
                                               const float* __restrict__ lam0,
                                               float* __restrict__ out) {
  __shared__ float pj[2 * Hd];
  __shared__ float hj[Hd];
  __shared__ float red[256];
  const int t = threadIdx.x;
  {
    float v = pooled[t] * (1.0f / (float)Nn);
    pj[t] = v;
    out[1537 + t] = v;  // hE: 1537..1664, hI: 1665..1792
  }
  __syncthreads();
  if (t < Hd) {
    float s = fb[t];
    for (int k = 0; k < 2 * Hd; ++k) s = fmaf(pj[k], fw[k * Hd + t], s);
    hj[t] = fmaxf(s, 0.0f);
  }
  __syncthreads();
  for (int o = t; o < 2 * Gd; o += 256) {
    float s = hb[o];
    for (int j = 0; j < Hd; ++j) s = fmaf(hj[j], hw[j * (2 * Gd) + o], s);
    out[o] = s;
  }
  float part = 0.0f;
  for (int gi = t; gi < Gd; gi += 256) {
    float K = expf(logK[gi]);
    out[1025 + gi] = K;
    part += K * tau[gi] / tau_max[gi];
  }
  red[t] = part;
  __syncthreads();
  for (int off = 128; off; off >>= 1) {
    if (t < off) red[t] += red[t + off];
    __syncthreads();
  }
  if (t == 0) out[1024] = fabsf(lam0[0]) - red[0];
}

// ---------------- host ----------------
extern "C" void kernel_launch(void* const* d_in, const int* in_sizes, int n_in,
                              void* d_out, int out_size, void* d_ws, size_t ws_size,
                              hipStream_t stream) {
  (void)in_sizes; (void)n_in; (void)out_size; (void)ws_size;
  const float* energy_x  = (const float*)d_in[0];
  const float* comm_x    = (const float*)d_in[1];
  const float* tau       = (const float*)d_in[2];
  const float* tau_max   = (const float*)d_in[3];
  const float* lam0      = (const float*)d_in[4];
  const float* e_proj_w  = (const float*)d_in[5];
  const float* e_proj_b  = (const float*)d_in[6];
  const float* c_proj_w  = (const float*)d_in[7];
  const float* c_proj_b  = (const float*)d_in[8];
  const float* e_self_w  = (const float*)d_in[9];
  const float* e_self_b  = (const float*)d_in[10];
  const float* e_neigh_w = (const float*)d_in[11];
  const float* e_neigh_b = (const float*)d_in[12];
  const float* e_ln_g    = (const float*)d_in[13];
  const float* e_ln_b    = (const float*)d_in[14];
  const float* c_self_w  = (const float*)d_in[15];
  const float* c_self_b  = (const float*)d_in[16];
  const float* c_neigh_w = (const float*)d_in[17];
  const float* c_neigh_b = (const float*)d_in[18];
  const float* c_ln_g    = (const float*)d_in[19];
  const float* c_ln_b    = (const float*)d_in[20];
  const float* fusion_w  = (const float*)d_in[21];
  const float* fusion_b  = (const float*)d_in[22];
  const float* head_w    = (const float*)d_in[23];
  const float* head_b    = (const float*)d_in[24];
  const float* log_K     = (const float*)d_in[25];
  const int*   e_edges   = (const int*)d_in[26];
  const int*   c_edges   = (const int*)d_in[27];

  float* ws     = (float*)d_ws;
  const size_t NH = (size_t)Nn * Hd;
  float* h      = ws;            // [N,H] activations
  float* tmp    = ws + NH;       // [N,H] GEMM output (neigh then self)
  float* agg    = ws + 2 * NH;   // [N,H] scatter accumulator
  float* cnt    = ws + 3 * NH;   // [N]   degree counts
  float* pooled = cnt + Nn;      // [256] pooled sums (E | I)

  const int gNH   = (int)((NH + 255) / 256);
  const int gGemm = (Nn + 63) / 64;
  const int gUpd  = (Nn + 7) / 8;
  const int gScat = (Ee * 32 + 255) / 256;
  const int gCnt  = (Ee + 255) / 256;
  const int gPool = (Nn + 511) / 512;

  k_zero<<<1, 256, 0, stream>>>(pooled, 256);

  for (int d = 0; d < 2; ++d) {
    const float* x  = d ? comm_x : energy_x;
    const int indim = d ? 3 : 5;
    const float* pw = d ? c_proj_w : e_proj_w;
    const float* pb = d ? c_proj_b : e_proj_b;
    const float* sw = d ? c_self_w : e_self_w;
    const float* sb = d ? c_self_b : e_self_b;
    const float* nw = d ? c_neigh_w : e_neigh_w;
    const float* nb = d ? c_neigh_b : e_neigh_b;
    const float* lg = d ? c_ln_g : e_ln_g;
    const float* lb = d ? c_ln_b : e_ln_b;
    const int* row  = d ? c_edges : e_edges;
    const int* col  = row + Ee;

    k_proj<<<gNH, 256, 0, stream>>>(x, pw, pb, h, indim);
    k_zero<<<(Nn + 255) / 256, 256, 0, stream>>>(cnt, Nn);
    k_count<<<gCnt, 256, 0, stream>>>(row, cnt);

    for (int l = 0; l < Ld; ++l) {
      // y = h @ neigh_w + neigh_b  (transform-then-gather: (x@W)[col] == x[col]@W)
      k_gemm<<<gGemm, 256, 0, stream>>>(h, nw + (size_t)l * Hd * Hd, nb + l * Hd, tmp);
      k_zero<<<gNH, 256, 0, stream>>>(agg, (int)NH);
      k_scatter<<<gScat, 256, 0, stream>>>(tmp, row, col, agg);
      // self transform (reuses tmp)
      k_gemm<<<gGemm, 256, 0, stream>>>(h, sw + (size_t)l * Hd * Hd, sb + l * Hd, tmp);
      k_update<<<gUpd, 256, 0, stream>>>(h, tmp, agg, cnt, lg + l * Hd, lb + l * Hd);
    }
    k_pool<<<gPool, 128, 0, stream>>>(h, pooled + d * Hd);
  }

  k_final<<<1, 256, 0, stream>>>(pooled, fusion_w, fusion_b, head_w, head_b,
                                 log_K, tau, tau_max, lam0, (float*)d_out);
}